// GroupWiseTemporalAttention_18236431139047
// MI455X (gfx1250) — compile-verified
//
#include <hip/hip_runtime.h>

// GroupWiseTemporalAttention on MI455X (gfx1250).
// Softmax over a single key == identity -> out == v. Whole op is one affine map:
//   y = x @ W_eff^T + b_eff,  W_eff = proj_w @ BlockDiag4(qkv_w[8:12,:]) (768x768)
//   b_eff = W_eff @ pe + proj_w @ rep(qkv_b[8:12]) + proj_b
// Single fp32 GEMM (M=50176, N=768, K=768) via v_wmma_f32_16x16x4_f32.
// Roofline: 59.2 GFLOP, 310 MB HBM traffic (AI~190) -> ~13us @ 23.3 TB/s; HBM-bound,
// so full-precision fp32 WMMA is free. Secondary constraint: W_eff (2.25MB, L2-resident)
// is re-streamed by every M panel; 64-row panels cut L2->WGP W traffic to
// 784 * 2.25MB = 1.76GB (vs 7.1GB at 16 rows), keeping us HBM-bound, not L2-bound.
// x-panel staging uses GLOBAL_LOAD_ASYNC_TO_LDS_B128 (ASYNCcnt path, no VGPR roundtrip).

typedef __attribute__((ext_vector_type(2))) float v2f;
typedef __attribute__((ext_vector_type(8))) float v8f;
typedef __attribute__((ext_vector_type(4))) int   v4i;

#define CD        768
#define KSTEPS    192        // 768 / 4
#define NT_WAVE   6          // 6 * 16 = 96 cols per wave; 8 waves cover N=768
#define MT_WAVE   4          // 4 * 16 = 64 rows per block (all waves share them)
#define MROWS     64
#define LDS_PITCH 772        // 768 + 4 pad -> conflict-free LDS A-fragment reads
#define LDS_BYTES (MROWS * LDS_PITCH * 4)   // 197632 B < 320KB/WGP

#if defined(__gfx1250__) && __has_builtin(__builtin_amdgcn_global_load_async_to_lds_b128)
#define USE_ASYNC_LDS 1
#endif

typedef __attribute__((address_space(1))) v4i* g_v4i_p;
typedef __attribute__((address_space(3))) v4i* l_v4i_p;

__device__ __forceinline__ void wait_async_zero() {
#if __has_builtin(__builtin_amdgcn_s_wait_asynccnt)
    __builtin_amdgcn_s_wait_asynccnt(0);
#else
    asm volatile("s_wait_asynccnt 0" ::: "memory");
#endif
}

// ---- Prologue A: fused weight in WMMA-B swizzled layout -------------------
// wsB element id = (t*192 + s)*32 + lane, two floats per id:
//   wsB[id*2+u] = W_eff[ n = 16t + lane%16 , k = 4s + (lane>=16?2:0) + u ]
__global__ void __launch_bounds__(256) build_weff(
    const float* __restrict__ proj_w,   // [768,768]
    const float* __restrict__ qkv_w,    // [12,4]
    float* __restrict__ wsB)
{
    int id   = blockIdx.x * 256 + threadIdx.x;      // 48*192*32 = 294912 total
    int lane = id & 31;
    int s    = (id >> 5) % KSTEPS;
    int t    = id / (32 * KSTEPS);
    int n    = t * 16 + (lane & 15);
    int kb   = s * 4 + ((lane >= 16) ? 2 : 0);
    const float* pw = proj_w + (size_t)n * CD + s * 4;
    float w0 = 0.f, w1 = 0.f;
    #pragma unroll
    for (int j = 0; j < 4; ++j) {
        float p = pw[j];                              // proj_w[n, 4s+j]
        w0 += p * qkv_w[(8 + j) * 4 + (kb & 3)];      // v_w[j, k%4]
        w1 += p * qkv_w[(8 + j) * 4 + ((kb + 1) & 3)];
    }
    wsB[(size_t)id * 2 + 0] = w0;
    wsB[(size_t)id * 2 + 1] = w1;
}

// ---- Prologue B: fused bias ----------------------------------------------
__global__ void __launch_bounds__(256) build_bias(
    const float* __restrict__ proj_w,
    const float* __restrict__ proj_b,
    const float* __restrict__ qkv_w,
    const float* __restrict__ qkv_b,
    const float* __restrict__ pe,
    float* __restrict__ beff)
{
    int n = blockIdx.x * 256 + threadIdx.x;
    if (n >= CD) return;
    float acc = proj_b[n];
    for (int c = 0; c < CD; ++c) {
        int j = c & 3, base = c & ~3;
        const float* qw = qkv_w + (8 + j) * 4;
        float ve = qkv_b[8 + j]
                 + qw[0] * pe[base] + qw[1] * pe[base + 1]
                 + qw[2] * pe[base + 2] + qw[3] * pe[base + 3];
        acc += proj_w[(size_t)n * CD + c] * ve;       // proj_w @ v(pe)
    }
    beff[n] = acc;
}

// ---- Main GEMM: y = x @ W_eff^T + b_eff ----------------------------------
// Block: 256 threads (8 wave32), 64M x 768N tile. Wave w owns N cols [96w, 96w+96).
// Each wave: 4 M-tiles x 6 N-tiles of 16x16, K looped in steps of 4.
// Per K-step: 4 LDS A-frag loads + 6 global B-frag loads -> 24 WMMAs
// (each B fragment reused 4x from registers).
__global__ void __launch_bounds__(256) wta_gemm(
    const float* __restrict__ x,     // [50176, 768]
    const float* __restrict__ wsB,   // swizzled W_eff
    const float* __restrict__ beff,  // [768]
    float* __restrict__ out)         // [50176, 768]
{
    extern __shared__ float xs[];    // MROWS * LDS_PITCH floats

    const int g    = blockIdx.x;     // 64-row M panel
    const int tid  = threadIdx.x;
    const int lane = tid & 31;
    const int wave = tid >> 5;

    // ---- Stage 64 x 768 x-panel into LDS: 12288 float4 chunks, 48/thread.
    const float4* xg4 = (const float4*)(x + (size_t)g * MROWS * CD);
    #pragma unroll 4
    for (int i = 0; i < 48; ++i) {
        int c    = tid + i * 256;            // chunk id 0..12287
        int row  = c / 192;                  // 192 float4 per row
        int col4 = c % 192;
#ifdef USE_ASYNC_LDS
        v4i* gp = (v4i*)(xg4 + row * 192 + col4);          // generic typed
        v4i* lp = (v4i*)&xs[row * LDS_PITCH + col4 * 4];   // generic typed
        __builtin_amdgcn_global_load_async_to_lds_b128(
            (g_v4i_p)gp, (l_v4i_p)lp, /*offset=*/0, /*cpol=*/0);
#else
        float4 v = xg4[row * 192 + col4];
        *(float4*)&xs[row * LDS_PITCH + col4 * 4] = v;
#endif
    }
#ifdef USE_ASYNC_LDS
    wait_async_zero();
#endif
    __syncthreads();

    // A fragment source (ISA 16x4 layout): lane%16 = M row within tile,
    // lanes>=16 take K+2; two consecutive K per v2f load.
    const int    kOff  = (lane >> 4) << 1;
    const float* aBase = &xs[(lane & 15) * LDS_PITCH + kOff];

    const v2f* Bp = (const v2f*)wsB;
    const v2f* bptr[NT_WAVE];
    #pragma unroll
    for (int t = 0; t < NT_WAVE; ++t)
        bptr[t] = Bp + (size_t)(wave * NT_WAVE + t) * KSTEPS * 32 + lane;

    v8f acc[MT_WAVE][NT_WAVE] = {};

    for (int s = 0; s < KSTEPS; ++s) {
        v2f a[MT_WAVE];
        #pragma unroll
        for (int m = 0; m < MT_WAVE; ++m)
            a[m] = *(const v2f*)(aBase + m * 16 * LDS_PITCH + 4 * s);
        #pragma unroll
        for (int t = 0; t < NT_WAVE; ++t) {
            v2f b = bptr[t][s * 32];
            #pragma unroll
            for (int m = 0; m < MT_WAVE; ++m)
                acc[m][t] = __builtin_amdgcn_wmma_f32_16x16x4_f32(
                    /*neg_a=*/false, a[m], /*neg_b=*/false, b,
                    /*c_mod=*/(short)0, acc[m][t],
                    /*reuse_a=*/false, /*reuse_b=*/false);
        }
    }

    // C/D layout: VGPR r -> row r (lanes 0-15) / row 8+r (lanes 16-31),
    // N = lane%16 within tile.
    const int colLocal = lane & 15;
    const int rowOff   = (lane >> 4) * 8;
    #pragma unroll
    for (int t = 0; t < NT_WAVE; ++t) {
        int   n    = wave * 96 + t * 16 + colLocal;
        float bias = beff[n];
        #pragma unroll
        for (int m = 0; m < MT_WAVE; ++m) {
            float* op = out + ((size_t)g * MROWS + m * 16 + rowOff) * CD + n;
            #pragma unroll
            for (int r = 0; r < 8; ++r)
                op[(size_t)r * CD] = acc[m][t][r] + bias;
        }
    }
}

extern "C" void kernel_launch(void* const* d_in, const int* in_sizes, int n_in,
                              void* d_out, int out_size, void* d_ws, size_t ws_size,
                              hipStream_t stream) {
    const float* x      = (const float*)d_in[0];   // [16,56,56,768]
    const float* qkv_w  = (const float*)d_in[1];   // [12,4]
    const float* qkv_b  = (const float*)d_in[2];   // [12]
    const float* proj_w = (const float*)d_in[3];   // [768,768]
    const float* proj_b = (const float*)d_in[4];   // [768]
    const float* pe     = (const float*)d_in[5];   // [1024]
    float* out  = (float*)d_out;
    float* wsB  = (float*)d_ws;                    // 768*768 floats
    float* beff = wsB + CD * CD;                   // + 768 floats

    // Allow >64KB dynamic LDS (gfx1250 WGP has 320KB). Not a stream op; safe
    // to call before capture-replayed launches.
    (void)hipFuncSetAttribute((const void*)wta_gemm,
                              hipFuncAttributeMaxDynamicSharedMemorySize,
                              LDS_BYTES);

    build_weff<<<dim3(48 * KSTEPS * 32 / 256), dim3(256), 0, stream>>>(proj_w, qkv_w, wsB);
    build_bias<<<dim3(3), dim3(256), 0, stream>>>(proj_w, proj_b, qkv_w, qkv_b, pe, beff);
    wta_gemm<<<dim3(50176 / MROWS), dim3(256), LDS_BYTES, stream>>>(x, wsB, beff, out);
}